// SparseGCN_64716567216341
// MI455X (gfx1250) — compile-verified
//
#include <hip/hip_runtime.h>
#include <math.h>

typedef float v2f __attribute__((ext_vector_type(2)));
typedef float v8f __attribute__((ext_vector_type(8)));

#if __has_builtin(__builtin_amdgcn_wmma_f32_16x16x4_f32)
#define HAVE_WMMA_F32 1
#endif

// ---------------------------------------------------------------------------
// 1) deg[i] = 1 (self-loop), zero the readout accumulator g[64]
__global__ void k_init(float* __restrict__ deg, float* __restrict__ g, int n) {
    int i = blockIdx.x * blockDim.x + threadIdx.x;
    if (i < n) deg[i] = 1.0f;
    if (i < 64) g[i] = 0.0f;
}

// 2) deg[dst[e]] += 1
__global__ void k_degcount(const int* __restrict__ dst, float* __restrict__ deg, int E) {
    int e = blockIdx.x * blockDim.x + threadIdx.x;
    if (e < E) atomicAdd(&deg[dst[e]], 1.0f);
}

// 3) dinv = rsqrt(deg)   (deg >= 1 always, no zero check needed)
__global__ void k_dinv(const float* __restrict__ deg, float* __restrict__ dinv, int n) {
    int i = blockIdx.x * blockDim.x + threadIdx.x;
    if (i < n) dinv[i] = rsqrtf(deg[i]);
}

// 4) h1 = x @ W1 ;  a1 = b1 + h1 * dinv^2  (self-loop term folded in)
//    one thread per (node, feature): idx = i*32 + j
__global__ void k_layer1(const float* __restrict__ x, const float* __restrict__ W1,
                         const float* __restrict__ b1, const float* __restrict__ dinv,
                         float* __restrict__ h1, float* __restrict__ a1, int n) {
    int idx = blockIdx.x * blockDim.x + threadIdx.x;
    int i = idx >> 5, j = idx & 31;
    if (i >= n) return;
    float x0 = x[i * 3 + 0], x1 = x[i * 3 + 1], x2 = x[i * 3 + 2];
    float v = fmaf(x0, W1[j], fmaf(x1, W1[32 + j], x2 * W1[64 + j]));
    h1[i * 32 + j] = v;
    float di = dinv[i];
    a1[i * 32 + j] = b1[j] + v * di * di;
}

// 5) a1[dst] += h1[src] * dinv[src]*dinv[dst]   (8 lanes x float4 per edge)
__global__ void k_scatter1(const int* __restrict__ src, const int* __restrict__ dst,
                           const float* __restrict__ dinv, const float* __restrict__ h1,
                           float* __restrict__ a1, int E) {
    int t = blockIdx.x * blockDim.x + threadIdx.x;
    int e = t >> 3, f = (t & 7) * 4;
    if (e >= E) return;
    int s = src[e], d = dst[e];
    float nrm = dinv[s] * dinv[d];
    float4 hv = *(const float4*)(h1 + (size_t)s * 32 + f);
    float* o = a1 + (size_t)d * 32 + f;
    atomicAdd(o + 0, hv.x * nrm);
    atomicAdd(o + 1, hv.y * nrm);
    atomicAdd(o + 2, hv.z * nrm);
    atomicAdd(o + 3, hv.w * nrm);
}

// 6) h2 = relu(a1) @ W2 ;  a2 = b2 + h2 * dinv^2
//    One wave per 16-node tile; 8 K-steps of V_WMMA_F32_16X16X4_F32 x 4 N-tiles.
__global__ void __launch_bounds__(128)
k_layer2(const float* __restrict__ a1, const float* __restrict__ W2,
         const float* __restrict__ b2, const float* __restrict__ dinv,
         float* __restrict__ h2, float* __restrict__ a2, int n) {
    int wave = blockIdx.x * (blockDim.x >> 5) + (threadIdx.x >> 5);
    int lane = threadIdx.x & 31;
    int base = wave * 16;
    if (base >= n) return;          // wave-uniform: EXEC stays all-ones for WMMA
    int half = lane >> 4;           // 0: K even pair / rows 0-7 ; 1: K odd pair / rows 8-15
    int l = lane & 15;
#ifdef HAVE_WMMA_F32
    int node = base + l; if (node >= n) node = n - 1;   // safe clamp (n%16==0 here)
    const float* arow = a1 + (size_t)node * 32;
    v8f c0 = {0,0,0,0,0,0,0,0}, c1 = c0, c2 = c0, c3 = c0;
#pragma unroll
    for (int kk = 0; kk < 8; ++kk) {
        int k0 = kk * 4 + half * 2;
        v2f a;                                  // A: 16x4 tile, fused relu
        a.x = fmaxf(arow[k0], 0.0f);
        a.y = fmaxf(arow[k0 + 1], 0.0f);
        const float* w0 = W2 + (size_t)k0 * 64;
        const float* w1 = W2 + (size_t)(k0 + 1) * 64;
        v2f b0v; b0v.x = w0[l];      b0v.y = w1[l];
        v2f b1v; b1v.x = w0[16 + l]; b1v.y = w1[16 + l];
        v2f b2v; b2v.x = w0[32 + l]; b2v.y = w1[32 + l];
        v2f b3v; b3v.x = w0[48 + l]; b3v.y = w1[48 + l];
        c0 = __builtin_amdgcn_wmma_f32_16x16x4_f32(false, a, false, b0v, (short)0, c0, false, false);
        c1 = __builtin_amdgcn_wmma_f32_16x16x4_f32(false, a, false, b1v, (short)0, c1, false, false);
        c2 = __builtin_amdgcn_wmma_f32_16x16x4_f32(false, a, false, b2v, (short)0, c2, false, false);
        c3 = __builtin_amdgcn_wmma_f32_16x16x4_f32(false, a, false, b3v, (short)0, c3, false, false);
    }
#pragma unroll
    for (int r = 0; r < 8; ++r) {
        int m = base + half * 8 + r;            // C/D layout: vgpr r, lane half selects row block
        if (m >= n) continue;
        float di = dinv[m], d2 = di * di;
        size_t o = (size_t)m * 64;
        float v;
        v = c0[r]; h2[o +      l] = v; a2[o +      l] = b2[l]      + v * d2;
        v = c1[r]; h2[o + 16 + l] = v; a2[o + 16 + l] = b2[16 + l] + v * d2;
        v = c2[r]; h2[o + 32 + l] = v; a2[o + 32 + l] = b2[32 + l] + v * d2;
        v = c3[r]; h2[o + 48 + l] = v; a2[o + 48 + l] = b2[48 + l] + v * d2;
    }
#else
    // scalar fallback (should not be taken per probe data)
    int node = base + l; if (node >= n) node = n - 1;
    for (int j = half * 32; j < half * 32 + 32; ++j) {
        float acc = 0.0f;
        for (int k = 0; k < 32; ++k)
            acc = fmaf(fmaxf(a1[(size_t)node * 32 + k], 0.0f), W2[(size_t)k * 64 + j], acc);
        float di = dinv[node];
        h2[(size_t)node * 64 + j] = acc;
        a2[(size_t)node * 64 + j] = b2[j] + acc * di * di;
    }
#endif
}

// 7) a2[dst] += h2[src] * dinv[src]*dinv[dst]   (16 lanes x float4 per edge)
__global__ void k_scatter2(const int* __restrict__ src, const int* __restrict__ dst,
                           const float* __restrict__ dinv, const float* __restrict__ h2,
                           float* __restrict__ a2, int E) {
    int t = blockIdx.x * blockDim.x + threadIdx.x;
    int e = t >> 4, f = (t & 15) * 4;
    if (e >= E) return;
    int s = src[e], d = dst[e];
    float nrm = dinv[s] * dinv[d];
    float4 hv = *(const float4*)(h2 + (size_t)s * 64 + f);
    float* o = a2 + (size_t)d * 64 + f;
    atomicAdd(o + 0, hv.x * nrm);
    atomicAdd(o + 1, hv.y * nrm);
    atomicAdd(o + 2, hv.z * nrm);
    atomicAdd(o + 3, hv.w * nrm);
}

// 8) g[j] = sum_i relu(a2[i,j])
__global__ void k_readout(const float* __restrict__ a2, float* __restrict__ g, int n) {
    int j = threadIdx.x & 63;
    int rowsPerBlk = blockDim.x >> 6;
    int i0 = blockIdx.x * rowsPerBlk + (threadIdx.x >> 6);
    int stride = gridDim.x * rowsPerBlk;
    float acc = 0.0f;
    for (int i = i0; i < n; i += stride)
        acc += fmaxf(a2[(size_t)i * 64 + j], 0.0f);
    atomicAdd(&g[j], acc);
}

// 9) y = sigmoid(g . Wfc + bfc)   (single wave32)
__global__ void k_final(const float* __restrict__ g, const float* __restrict__ Wfc,
                        const float* __restrict__ bfc, float* __restrict__ out) {
    int l = threadIdx.x;
    float s = g[l] * Wfc[l] + g[l + 32] * Wfc[l + 32];
    for (int off = 16; off > 0; off >>= 1) s += __shfl_down(s, off, 32);
    if (l == 0) out[0] = 1.0f / (1.0f + expf(-(s + bfc[0])));
}

// ---------------------------------------------------------------------------
extern "C" void kernel_launch(void* const* d_in, const int* in_sizes, int n_in,
                              void* d_out, int out_size, void* d_ws, size_t ws_size,
                              hipStream_t stream) {
    const float* x   = (const float*)d_in[0];
    const int*   ei  = (const int*)d_in[1];    // JAX x64-off: int64 request -> int32
    const float* W1  = (const float*)d_in[2];
    const float* b1  = (const float*)d_in[3];
    const float* W2  = (const float*)d_in[4];
    const float* b2  = (const float*)d_in[5];
    const float* Wfc = (const float*)d_in[6];
    const float* bfc = (const float*)d_in[7];
    float* out = (float*)d_out;

    int n = in_sizes[0] / 3;
    int E = in_sizes[1] / 2;
    const int* src = ei;
    const int* dst = ei + E;

    // workspace layout (floats): deg | dinv | h1(32n) | a1(32n) | h2(64n) | a2(64n) | g(64)
    float* ws   = (float*)d_ws;
    float* deg  = ws;
    float* dinv = ws + (size_t)n;
    float* h1   = ws + (size_t)2 * n;
    float* a1   = ws + (size_t)34 * n;
    float* h2   = ws + (size_t)66 * n;
    float* a2   = ws + (size_t)130 * n;
    float* g    = ws + (size_t)194 * n;

    const int B = 256;
    k_init    <<<(n + B - 1) / B, B, 0, stream>>>(deg, g, n);
    k_degcount<<<(E + B - 1) / B, B, 0, stream>>>(dst, deg, E);
    k_dinv    <<<(n + B - 1) / B, B, 0, stream>>>(deg, dinv, n);

    long long t1 = (long long)n * 32;
    k_layer1  <<<(int)((t1 + B - 1) / B), B, 0, stream>>>(x, W1, b1, dinv, h1, a1, n);

    long long t2 = (long long)E * 8;
    k_scatter1<<<(int)((t2 + B - 1) / B), B, 0, stream>>>(src, dst, dinv, h1, a1, E);

    int waves = (n + 15) / 16;                       // 3125 tiles
    k_layer2  <<<(waves + 3) / 4, 128, 0, stream>>>(a1, W2, b2, dinv, h2, a2, n);

    long long t3 = (long long)E * 16;
    k_scatter2<<<(int)((t3 + B - 1) / B), B, 0, stream>>>(src, dst, dinv, h2, a2, E);

    k_readout <<<512, 256, 0, stream>>>(a2, g, n);
    k_final   <<<1, 32, 0, stream>>>(g, Wfc, bfc, out);
}